// LSTM_30021821399394
// MI455X (gfx1250) — compile-verified
//
#include <hip/hip_runtime.h>

// ---------------------------------------------------------------------------
// Fused LSTM (B=4096, T=256, F=18, H=64) + linear head (O=15), fp32.
// One workgroup = 16 batch rows, 4 waves (wave32). Wave j owns h-columns
// [16j, 16j+16). Per timestep each wave computes its 16x16 tile of all four
// gates with V_WMMA_F32_16X16X4_F32 chains (full fp32 precision), updates its
// c/h fragment, and shares h through double-buffered LDS for the next step.
// Weight fragments live in VGPRs for the whole time loop.
// ---------------------------------------------------------------------------

typedef float v2f __attribute__((ext_vector_type(2)));
typedef float v8f __attribute__((ext_vector_type(8)));

#define TS    256   // timesteps
#define FIN   18    // input features (padded to 20 for K=4 steps)
#define HID   64    // hidden size
#define G4    256   // 4*HID
#define NOUT  15    // classifier outputs
#define BM    16    // batch rows per workgroup
#define HPAD  (HID + 4)   // LDS row stride (68 dwords) -> bank-conflict-free frags

__device__ __forceinline__ float fast_sig(float x) {
    // 1 / (1 + e^-x)
    return __builtin_amdgcn_rcpf(1.0f + __expf(-x));
}
__device__ __forceinline__ float fast_tanh(float x) {
    // 1 - 2 / (e^(2x) + 1)   (correct limits at +-inf)
    return 1.0f - 2.0f * __builtin_amdgcn_rcpf(1.0f + __expf(2.0f * x));
}

__global__ __launch_bounds__(128)
void lstm_fused_kernel(const float* __restrict__ X,
                       const float* __restrict__ W_ih,
                       const float* __restrict__ W_hh,
                       const float* __restrict__ b_ih,
                       const float* __restrict__ b_hh,
                       const float* __restrict__ W_cls,
                       const float* __restrict__ b_cls,
                       float* __restrict__ out)
{
    __shared__ float hbuf[2][BM][HPAD];   // double-buffered hidden state tile

    const int tid  = threadIdx.x;
    const int wave = tid >> 5;            // j: 16-column slice of H
    const int lane = tid & 31;
    const int m    = lane & 15;           // A-row / B-column within tile
    const int g2   = (lane >> 4) << 1;    // K-pair selector: 0 or 2
    const int b0   = blockIdx.x * BM;     // first batch row of this block

    // ---- hoist weight B-fragments into registers (loop-invariant) --------
    // B layout (4x16 f32): lane holds B[k][n], n = lane&15, k = 2*(lane>>4)+v,
    // i.e. W[n_global][4*kstep + g2 + {0,1}] -> contiguous float2.
    v2f  bh[4][16];   // W_hh^T fragments: gate q, K-step k  (K = 64)
    v2f  bx[4][5];    // W_ih^T fragments: gate q, K-step k  (K = 18 -> 20)
    float bias[4];

#pragma unroll
    for (int q = 0; q < 4; ++q) {
        const int n = q * HID + wave * 16 + m;        // gate column index
#pragma unroll
        for (int k = 0; k < 16; ++k)
            bh[q][k] = *(const v2f*)(W_hh + n * HID + 4 * k + g2);
#pragma unroll
        for (int k = 0; k < 5; ++k) {
            const int col = 4 * k + g2;
            v2f w = {0.0f, 0.0f};
            if (col + 1 < FIN) w = *(const v2f*)(W_ih + n * FIN + col);
            bx[q][k] = w;
        }
        bias[q] = b_ih[n] + b_hh[n];
    }

    // ---- zero h buffer 0 ---------------------------------------------------
    for (int i = tid; i < BM * HPAD; i += 128)
        ((float*)hbuf[0])[i] = 0.0f;

    v8f c = {};                            // cell-state fragment (16x16 tile)
    int cur = 0;
    const float* xrow = X + (long)(b0 + m) * TS * FIN;   // this lane's A-row

    for (int t = 0; t < TS; ++t) {
        __syncthreads();                   // h[cur] (and init) visible

        // ---- A-fragments ---------------------------------------------------
        // A layout (16x4 f32): lane holds A[m][4*kstep + g2 + {0,1}].
        v2f ax[5];
#pragma unroll
        for (int k = 0; k < 5; ++k) {
            const int col = 4 * k + g2;
            v2f a = {0.0f, 0.0f};
            if (col + 1 < FIN) a = *(const v2f*)(xrow + t * FIN + col);
            ax[k] = a;
        }
        v2f ah[16];
#pragma unroll
        for (int k = 0; k < 16; ++k)
            ah[k] = *(const v2f*)&hbuf[cur][m][4 * k + g2];

        // ---- gates = X_t @ W_ih^T + h @ W_hh^T + (b_ih + b_hh) ------------
        v8f acc[4];
#pragma unroll
        for (int q = 0; q < 4; ++q) {
            v8f a;
#pragma unroll
            for (int e = 0; e < 8; ++e) a[e] = bias[q];   // column bias splat
#pragma unroll
            for (int k = 0; k < 5; ++k)
                a = __builtin_amdgcn_wmma_f32_16x16x4_f32(
                        false, ax[k], false, bx[q][k], (short)0, a, false, false);
#pragma unroll
            for (int k = 0; k < 16; ++k)
                a = __builtin_amdgcn_wmma_f32_16x16x4_f32(
                        false, ah[k], false, bh[q][k], (short)0, a, false, false);
            acc[q] = a;
        }

        // ---- elementwise LSTM cell update (gate order: i, f, g, o) --------
        v8f h;
#pragma unroll
        for (int e = 0; e < 8; ++e) {
            const float ig = fast_sig (acc[0][e]);
            const float fg = fast_sig (acc[1][e]);
            const float gg = fast_tanh(acc[2][e]);
            const float og = fast_sig (acc[3][e]);
            const float cn = fg * c[e] + ig * gg;
            c[e] = cn;
            h[e] = og * fast_tanh(cn);
        }

        // ---- publish h tile to the other buffer ---------------------------
        // C/D layout: VGPR e -> row e + 8*(lane>>4), column = lane&15.
        const int nxt  = cur ^ 1;
        const int row0 = (lane >> 4) * 8;
#pragma unroll
        for (int e = 0; e < 8; ++e)
            hbuf[nxt][row0 + e][wave * 16 + m] = h[e];
        cur = nxt;
    }

    __syncthreads();                       // final h visible to everyone

    // ---- classifier: out = h_last @ W_cls^T + b_cls  (tiny: 16x15) --------
    for (int idx = tid; idx < BM * NOUT; idx += 128) {
        const int r = idx / NOUT;
        const int o = idx % NOUT;
        float s = b_cls[o];
#pragma unroll 8
        for (int k = 0; k < HID; ++k)
            s += hbuf[cur][r][k] * W_cls[o * HID + k];
        out[(b0 + r) * NOUT + o] = s;
    }
}

extern "C" void kernel_launch(void* const* d_in, const int* in_sizes, int n_in,
                              void* d_out, int out_size, void* d_ws, size_t ws_size,
                              hipStream_t stream) {
    const float* X     = (const float*)d_in[0];
    const float* W_ih  = (const float*)d_in[1];
    const float* W_hh  = (const float*)d_in[2];
    const float* b_ih  = (const float*)d_in[3];
    const float* b_hh  = (const float*)d_in[4];
    const float* W_cls = (const float*)d_in[5];
    const float* b_cls = (const float*)d_in[6];

    const int B = in_sizes[0] / (TS * FIN);      // 4096
    dim3 grid(B / BM);                            // 256 workgroups
    dim3 block(128);                              // 4 wave32 waves

    lstm_fused_kernel<<<grid, block, 0, stream>>>(
        X, W_ih, W_hh, b_ih, b_hh, W_cls, b_cls, (float*)d_out);
}